// LSH_9302899163580
// MI455X (gfx1250) — compile-verified
//
#include <hip/hip_runtime.h>
#include <hip/hip_bf16.h>
#include <stdint.h>

// -------- hash table config --------
#define HBITS 19u
#define HSIZE (1u << HBITS)            // 524288 slots >= 2x max distinct keys (262144)
#define EMPTY_KEY ((int)0x80000000)    // |v| << 2^31, never collides with a real key

typedef float v2f __attribute__((ext_vector_type(2)));
typedef float v8f __attribute__((ext_vector_type(8)));

// ---------------------------------------------------------------------------
// K0: v[i] = (int)trunc(dot(weights, x[i,:])) via V_WMMA_F32_16X16X4_F32.
// One wave32 handles 16 rows. A = weights chunk replicated over all 16 M rows
// (documented A 16x4 f32 layout: lane<16 -> K=0/1 in elem0/1, lane>=16 -> K=2/3).
// B[k][n] = x[n][k0+k] with N = lane%16, K = (lane/16)*2 + elem (mirrors the
// ISA's documented B striping). All 16 rows of C are identical; lane L's C[0]
// holds the dot for column N = L%16.
// ---------------------------------------------------------------------------
__global__ void k_hash_wmma(const float* __restrict__ x, const float* __restrict__ w,
                            int* __restrict__ v, int B, int D) {
  const int lane = threadIdx.x & 31;
  const int wave = (blockIdx.x * blockDim.x + threadIdx.x) >> 5;
  const int rowBase = wave * 16;
  if (rowBase >= B) return;                    // wave-uniform (B % 16 == 0)
  const int n  = rowBase + (lane & 15);        // column owned by this lane
  const int kh = (lane >> 4) << 1;             // 0 for lanes 0-15, 2 for 16-31

  const float* __restrict__ xr = x + (size_t)n * D;
  v8f c = {};
  for (int k0 = 0; k0 < D; k0 += 4) {
    v2f a, b;
    a.x = w[k0 + kh];      a.y = w[k0 + kh + 1];
    b.x = xr[k0 + kh];     b.y = xr[k0 + kh + 1];
    c = __builtin_amdgcn_wmma_f32_16x16x4_f32(false, a, false, b, (short)0, c,
                                              false, false);
  }
  if (lane < 16) v[rowBase + lane] = (int)truncf(c[0]);
}

// K1: reset hash table + counters (runs every call; graph-replay safe)
__global__ void k_init(int* __restrict__ hkeys, int* __restrict__ hcount,
                       int* __restrict__ hfill, int* __restrict__ counters) {
  unsigned i = blockIdx.x * blockDim.x + threadIdx.x;
  if (i < HSIZE) { hkeys[i] = EMPTY_KEY; hcount[i] = 0; hfill[i] = 0; }
  if (i < 2) counters[i] = 0;   // [0]=nbuckets, [1]=member alloc cursor
}

__device__ __forceinline__ unsigned hash_key(int key) {
  return ((unsigned)key * 2654435761u) >> (32u - HBITS);
}

// K2: insert each row's key; record its slot and bump the slot count
__global__ void k_insert(const int* __restrict__ v, int* __restrict__ hkeys,
                         int* __restrict__ hcount, int* __restrict__ rowslot, int B) {
  int i = blockIdx.x * blockDim.x + threadIdx.x;
  if (i >= B) return;
  const int key = v[i];
  unsigned h = hash_key(key) & (HSIZE - 1u);
  for (;;) {
    int old = atomicCAS(&hkeys[h], EMPTY_KEY, key);
    if (old == EMPTY_KEY || old == key) {
      rowslot[i] = (int)h;
      atomicAdd(&hcount[h], 1);
      return;
    }
    h = (h + 1u) & (HSIZE - 1u);
  }
}

// K3: allocate a contiguous member segment per non-empty slot; build compact list
__global__ void k_offsets(const int* __restrict__ hcount, int* __restrict__ hoffset,
                          int* __restrict__ nonempty, int* __restrict__ counters) {
  unsigned s = blockIdx.x * blockDim.x + threadIdx.x;
  if (s >= HSIZE) return;
  int cnt = hcount[s];
  if (cnt > 0) {
    hoffset[s] = atomicAdd(&counters[1], cnt);
    int b = atomicAdd(&counters[0], 1);
    nonempty[b] = (int)s;
  }
}

// K4: scatter row ids into member segments (arrival order is arbitrary here)
__global__ void k_fill(const int* __restrict__ rowslot, const int* __restrict__ hoffset,
                       int* __restrict__ hfill, int* __restrict__ members, int B) {
  int i = blockIdx.x * blockDim.x + threadIdx.x;
  if (i >= B) return;
  int s = rowslot[i];
  int pos = atomicAdd(&hfill[s], 1);
  members[hoffset[s] + pos] = i;
}

// K5: rank of each row within its bucket = #{members with smaller row index}
// (order-independent -> deterministic even though K4's layout is not)
__global__ void k_rank(const int* __restrict__ rowslot, const int* __restrict__ hoffset,
                       const int* __restrict__ hcount, const int* __restrict__ members,
                       int* __restrict__ rowrank, int B) {
  int i = blockIdx.x * blockDim.x + threadIdx.x;
  if (i >= B) return;
  int s = rowslot[i], off = hoffset[s], k = hcount[s];
  int r = 0;
  for (int t = 0; t < k; ++t) r += (members[off + t] < i) ? 1 : 0;
  rowrank[i] = r;
}

// K6: rewrite member segments in row-index order (ranks form a permutation)
__global__ void k_reorder(const int* __restrict__ rowslot, const int* __restrict__ hoffset,
                          const int* __restrict__ rowrank, int* __restrict__ members, int B) {
  int i = blockIdx.x * blockDim.x + threadIdx.x;
  if (i >= B) return;
  members[hoffset[rowslot[i]] + rowrank[i]] = i;
}

// K7: one workgroup (D=128 threads, thread = one dim) per non-empty bucket.
// Exact sequential EMA in sorted row order, then scatter trunc(acc) to members.
__global__ void k_bucket(const float* __restrict__ x, const int* __restrict__ members,
                         const int* __restrict__ hoffset, const int* __restrict__ hcount,
                         const int* __restrict__ nonempty, const int* __restrict__ counters,
                         float* __restrict__ out, int D) {
  int b = blockIdx.x;
  if (b >= counters[0]) return;
  int s = nonempty[b], off = hoffset[s], k = hcount[s];
  int d = threadIdx.x;
  float acc = x[(size_t)members[off] * D + d];
  for (int r = 1; r < k; ++r)
    acc = (acc + x[(size_t)members[off + r] * D + d]) * 0.5f;
  float res = truncf(acc);
  for (int r = 0; r < k; ++r)
    out[(size_t)members[off + r] * D + d] = res;
}

extern "C" void kernel_launch(void* const* d_in, const int* in_sizes, int n_in,
                              void* d_out, int out_size, void* d_ws, size_t ws_size,
                              hipStream_t stream) {
  const float* x = (const float*)d_in[0];
  const float* w = (const float*)d_in[1];
  float* out = (float*)d_out;
  const int D = in_sizes[1];            // 128
  const int B = in_sizes[0] / D;        // 262144

  // scratch layout (ints): ~13.6 MB total
  int* ws       = (int*)d_ws;
  int* v        = ws;                   // B
  int* rowslot  = v        + B;         // B
  int* rowrank  = rowslot  + B;         // B
  int* members  = rowrank  + B;         // B
  int* hkeys    = members  + B;         // HSIZE
  int* hcount   = hkeys    + HSIZE;     // HSIZE
  int* hoffset  = hcount   + HSIZE;     // HSIZE
  int* hfill    = hoffset  + HSIZE;     // HSIZE
  int* nonempty = hfill    + HSIZE;     // B
  int* counters = nonempty + B;         // 2

  const int rowsPerBlock = (256 / 32) * 16;   // 8 waves * 16 rows
  k_hash_wmma<<<(B + rowsPerBlock - 1) / rowsPerBlock, 256, 0, stream>>>(x, w, v, B, D);
  k_init   <<<(HSIZE + 255) / 256, 256, 0, stream>>>(hkeys, hcount, hfill, counters);
  k_insert <<<(B + 255) / 256, 256, 0, stream>>>(v, hkeys, hcount, rowslot, B);
  k_offsets<<<(HSIZE + 255) / 256, 256, 0, stream>>>(hcount, hoffset, nonempty, counters);
  k_fill   <<<(B + 255) / 256, 256, 0, stream>>>(rowslot, hoffset, hfill, members, B);
  k_rank   <<<(B + 255) / 256, 256, 0, stream>>>(rowslot, hoffset, hcount, members, rowrank, B);
  k_reorder<<<(B + 255) / 256, 256, 0, stream>>>(rowslot, hoffset, rowrank, members, B);
  k_bucket <<<B, D, 0, stream>>>(x, members, hoffset, hcount, nonempty, counters, out, D);
}